// HDCEncoder_27410481283307
// MI455X (gfx1250) — compile-verified
//
#include <hip/hip_runtime.h>
#include <hip/hip_bf16.h>

// HDC encoder = pure row gather: out[i, :] = vocab[tokens[i], :]
// N_TOKENS = 8192, DIMENSIONS = 10000, float32.
// Bandwidth-bound (~655 MB total @ 23.3 TB/s => ~28 us floor).
// Strategy: CDNA5 async copy engine, HBM -> LDS -> HBM, no VGPR round trip.
//   - loads:  global_load_async_to_lds_b128  (RT hint: gathered vocab rows
//             have repeats -> keep them resident in the 192 MB L2)
//   - stores: global_store_async_from_lds_b128 th:TH_STORE_NT (output is
//             write-once streaming; don't thrash L2 with it)

#define HDC_DIMS        10000
#define HDC_F4_PER_ROW  2500          // 10000 floats / 4 per b128
#define HDC_THREADS     256           // 8 wave32 waves per block

__global__ __launch_bounds__(HDC_THREADS)
void HDCEncoder_gather_async(const int* __restrict__ tokens,
                             const float* __restrict__ vocab,
                             float* __restrict__ out)
{
    // 40 KB LDS staging row (16B-aligned for b128 transfers); 320 KB/WGP
    // means several blocks can still co-reside per WGP.
    __shared__ __align__(16) float lds_row[HDC_DIMS];

    const unsigned tid = threadIdx.x;
    const int      row = blockIdx.x;

    // Uniform per block -> scalar load, row bases live in SGPR pairs.
    const int tok = tokens[row];
    const unsigned long long src_base =
        (unsigned long long)(const void*)(vocab + (unsigned long long)tok * HDC_DIMS);
    const unsigned long long dst_base =
        (unsigned long long)(void*)(out + (unsigned long long)row * HDC_DIMS);

    // LDS aperture lives entirely in addr[63:32]; low 32 bits of the generic
    // pointer are the LDS byte offset.
    const unsigned lds_base = (unsigned)(unsigned long long)(const void*)&lds_row[0];

    // Phase 1: async gather of the vocab row into LDS (GVS mode:
    // saddr = SGPR-pair row base, vaddr = per-lane byte offset).
    for (unsigned idx = tid; idx < HDC_F4_PER_ROW; idx += HDC_THREADS) {
        const unsigned byte_off = idx * 16u;
        const unsigned lds_addr = lds_base + byte_off;
        asm volatile(
            "global_load_async_to_lds_b128 %0, %1, %2"
            :
            : "v"(lds_addr), "v"(byte_off), "s"(src_base)
            : "memory");
    }

    // Wave-level wait: this wave's loads have landed in LDS. No workgroup
    // barrier needed -- each lane stores exactly the bytes it loaded.
    asm volatile("s_wait_asynccnt 0" ::: "memory");

    // Phase 2: async streaming store of the staged row to the output
    // (non-temporal: output is never re-read, keep L2 for vocab rows).
    for (unsigned idx = tid; idx < HDC_F4_PER_ROW; idx += HDC_THREADS) {
        const unsigned byte_off = idx * 16u;
        const unsigned lds_addr = lds_base + byte_off;
        asm volatile(
            "global_store_async_from_lds_b128 %0, %1, %2 th:TH_STORE_NT"
            :
            : "v"(byte_off), "v"(lds_addr), "s"(dst_base)
            : "memory");
    }

    // Ensure stores are retired before wave end (s_endpgm also wait-idles,
    // but be explicit).
    asm volatile("s_wait_asynccnt 0" ::: "memory");
}

extern "C" void kernel_launch(void* const* d_in, const int* in_sizes, int n_in,
                              void* d_out, int out_size, void* d_ws, size_t ws_size,
                              hipStream_t stream)
{
    const int*   tokens = (const int*)d_in[0];     // [N_TOKENS] int32
    const float* vocab  = (const float*)d_in[1];   // [VOCAB, DIMENSIONS] f32
    float*       out    = (float*)d_out;           // [N_TOKENS, DIMENSIONS] f32

    const int n_tokens = in_sizes[0];              // 8192

    HDCEncoder_gather_async<<<dim3(n_tokens), dim3(HDC_THREADS), 0, stream>>>(
        tokens, vocab, out);
}